// LinkPredictor_68221260529797
// MI455X (gfx1250) — compile-verified
//
#include <hip/hip_runtime.h>

typedef __attribute__((ext_vector_type(2))) float v2f;
typedef __attribute__((ext_vector_type(8))) float v8f;

#define NNODES 50000
#define FIN    256
#define FH     128

// ---------------------------------------------------------------- utilities
__global__ void zero_kernel(float* __restrict__ p, long n) {
    long i = (long)blockIdx.x * blockDim.x + threadIdx.x;
    if (i < n) p[i] = 0.0f;
}

__global__ void deg_kernel(const int* __restrict__ dst, float* __restrict__ deg, int E_) {
    int e = blockIdx.x * blockDim.x + threadIdx.x;
    if (e < E_) atomicAdd(&deg[dst[e]], 1.0f);
}

__global__ void dinv_kernel(float* __restrict__ d, int n) {
    int i = blockIdx.x * blockDim.x + threadIdx.x;
    if (i < n) d[i] = rsqrtf(d[i] + 1.0f);  // deg includes self-loop
}

// ------------------------------------------------------- fp32 WMMA GEMM
// C[M x 128] = A[M x K] @ B[K x 128], all fp32, via V_WMMA_F32_16X16X4_F32.
// One wave -> one 16x16 C tile. blockDim=256 (8 waves cover N=128).
// grid.x = M/16 (M multiple of 16). EXEC is all-ones throughout (WMMA req).
__global__ void gemm_wmma_f32(const float* __restrict__ A,
                              const float* __restrict__ B,
                              float* __restrict__ C, int K) {
    const int wave = threadIdx.x >> 5;      // N-tile index, 0..7
    const int lane = threadIdx.x & 31;
    const int r16  = lane & 15;             // M (for A) / N (for B,C) within tile
    const int half = lane >> 4;             // selects K pair {0,1} vs {2,3}
    const long aRow = (long)blockIdx.x * 16 + r16;
    const int  gCol = wave * 16 + r16;      // global column in [0,128)

    v8f acc = {};
    for (int k0 = 0; k0 < K; k0 += 4) {
        const int ka = k0 + 2 * half;
        // A 16x4 frag: VGPR0=K=ka, VGPR1=K=ka+1 (contiguous in row-major A)
        v2f a = *(const v2f*)(A + aRow * K + ka);
        // B 4x16 frag: row K striped across lanes within a VGPR
        v2f b;
        b.x = B[(long)ka * FH + gCol];
        b.y = B[(long)(ka + 1) * FH + gCol];
        acc = __builtin_amdgcn_wmma_f32_16x16x4_f32(
            /*neg_a=*/false, a, /*neg_b=*/false, b,
            /*c_mod=*/(short)0, acc, /*reuse_a=*/false, /*reuse_b=*/false);
    }
    // C layout: VGPR r -> row half*8 + r, col = lane%16
    const long base = ((long)blockIdx.x * 16 + half * 8) * FH + gCol;
#pragma unroll
    for (int r = 0; r < 8; ++r) C[base + (long)r * FH] = acc[r];
}

// ------------------------------------------------- edge propagation (scatter)
// agg[dst] += h[src] * (dinv[src]*dinv[dst]); one wave per edge, float4/lane.
__global__ void edge_scatter(const float* __restrict__ h,
                             const int* __restrict__ src,
                             const int* __restrict__ dst,
                             const float* __restrict__ dinv,
                             float* __restrict__ agg, int E_) {
    const long t = (long)blockIdx.x * blockDim.x + threadIdx.x;
    const long e = t >> 5;
    const int  lane = (int)(t & 31);
    if (e >= E_) return;
    const int s = src[e], d = dst[e];
    const float coef = dinv[s] * dinv[d];
    const float4 hv = *(const float4*)(h + (long)s * FH + lane * 4);
    float* ap = agg + (long)d * FH + lane * 4;
    atomicAdd(ap + 0, hv.x * coef);
    atomicAdd(ap + 1, hv.y * coef);
    atomicAdd(ap + 2, hv.z * coef);
    atomicAdd(ap + 3, hv.w * coef);
}

// ------------------------------------- z = [relu](agg + h*selfcoef + bias)
__global__ void combine_kernel(const float* __restrict__ agg,
                               const float* __restrict__ h,
                               const float* __restrict__ dinv,
                               const float* __restrict__ bias,
                               float* __restrict__ z, int do_relu) {
    const long t = (long)blockIdx.x * blockDim.x + threadIdx.x;
    if (t >= (long)NNODES * FH) return;
    const int node = (int)(t >> 7);
    const int f    = (int)(t & 127);
    float sc = dinv[node];
    sc *= sc;
    float v = agg[t] + h[t] * sc + bias[f];
    if (do_relu) v = fmaxf(v, 0.0f);
    z[t] = v;
}

// --------------------------------------------------------------- decoder
// out[l] = dot(concat(z[a], z[b], z[c]), Wt) + bt ; one wave per label.
__global__ void decode_kernel(const float* __restrict__ z,
                              const int* __restrict__ eli,
                              const float* __restrict__ Wt,
                              const float* __restrict__ bt,
                              float* __restrict__ out, int L_) {
    const long t = (long)blockIdx.x * blockDim.x + threadIdx.x;
    const long l = t >> 5;
    const int  lane = (int)(t & 31);
    if (l >= L_) return;
    const int n0 = eli[l];
    const int n1 = eli[(long)L_ + l];
    const int n2 = eli[2L * L_ + l];
    const int f = lane * 4;

    float acc = 0.0f;
    {
        float4 a = *(const float4*)(z + (long)n0 * FH + f);
        float4 w = *(const float4*)(Wt + f);
        acc += a.x * w.x + a.y * w.y + a.z * w.z + a.w * w.w;
    }
    {
        float4 a = *(const float4*)(z + (long)n1 * FH + f);
        float4 w = *(const float4*)(Wt + FH + f);
        acc += a.x * w.x + a.y * w.y + a.z * w.z + a.w * w.w;
    }
    {
        float4 a = *(const float4*)(z + (long)n2 * FH + f);
        float4 w = *(const float4*)(Wt + 2 * FH + f);
        acc += a.x * w.x + a.y * w.y + a.z * w.z + a.w * w.w;
    }
#pragma unroll
    for (int off = 16; off > 0; off >>= 1)
        acc += __shfl_down(acc, off, 32);
    if (lane == 0) out[l] = acc + bt[0];
}

// ---------------------------------------------------------------- launcher
extern "C" void kernel_launch(void* const* d_in, const int* in_sizes, int n_in,
                              void* d_out, int out_size, void* d_ws, size_t ws_size,
                              hipStream_t stream) {
    const float* x   = (const float*)d_in[0];
    const int*   ei  = (const int*)  d_in[1];   // (2, E)
    const int*   eli = (const int*)  d_in[2];   // (3, L)
    const float* W1  = (const float*)d_in[3];
    const float* b1  = (const float*)d_in[4];
    const float* W2  = (const float*)d_in[5];
    const float* b2  = (const float*)d_in[6];
    const float* Wt  = (const float*)d_in[7];
    const float* bt  = (const float*)d_in[8];
    float* out = (float*)d_out;

    const int E_ = in_sizes[1] / 2;
    const int L_ = in_sizes[2] / 3;
    const int* src = ei;
    const int* dst = ei + E_;

    // workspace carve-up: dinv[N] | h[N*128] | agg[N*128] | z[N*128]
    float* dinv = (float*)d_ws;
    float* h    = dinv + NNODES;
    float* agg  = h    + (long)NNODES * FH;
    float* z    = agg  + (long)NNODES * FH;

    const long NF = (long)NNODES * FH;
    const int TPB = 256;
    const int gNF   = (int)((NF + TPB - 1) / TPB);
    const int gN    = (NNODES + TPB - 1) / TPB;
    const int gE    = (E_ + TPB - 1) / TPB;
    const int gE32  = (int)(((long)E_ * 32 + TPB - 1) / TPB);
    const int gL32  = (int)(((long)L_ * 32 + TPB - 1) / TPB);
    const int gGemm = NNODES / 16;   // 3125 (exact)

    // degree -> dinv (dinv buffer doubles as deg accumulator)
    zero_kernel<<<gN, TPB, 0, stream>>>(dinv, NNODES);
    deg_kernel<<<gE, TPB, 0, stream>>>(dst, dinv, E_);
    dinv_kernel<<<gN, TPB, 0, stream>>>(dinv, NNODES);

    // ---- layer 1: h = x @ W1 ; agg = scatter ; z = relu(agg + h*sc + b1)
    gemm_wmma_f32<<<gGemm, TPB, 0, stream>>>(x, W1, h, FIN);
    zero_kernel<<<gNF, TPB, 0, stream>>>(agg, NF);
    edge_scatter<<<gE32, TPB, 0, stream>>>(h, src, dst, dinv, agg, E_);
    combine_kernel<<<gNF, TPB, 0, stream>>>(agg, h, dinv, b1, z, 1);

    // ---- layer 2: h = z @ W2 ; agg = scatter ; z = agg + h*sc + b2
    gemm_wmma_f32<<<gGemm, TPB, 0, stream>>>(z, W2, h, FH);
    zero_kernel<<<gNF, TPB, 0, stream>>>(agg, NF);
    edge_scatter<<<gE32, TPB, 0, stream>>>(h, src, dst, dinv, agg, E_);
    combine_kernel<<<gNF, TPB, 0, stream>>>(agg, h, dinv, b2, z, 0);

    // ---- decoder
    decode_kernel<<<gL32, TPB, 0, stream>>>(z, eli, Wt, bt, out, L_);
}